// FrameAveraging_11338713661595
// MI455X (gfx1250) — compile-verified
//
#include <hip/hip_runtime.h>
#include <math.h>

typedef float v2f __attribute__((ext_vector_type(2)));
typedef float v8f __attribute__((ext_vector_type(8)));

#define NB   256   // batches
#define NPTS 4096  // points per batch

// ---------------------------------------------------------------------------
// 3x3 symmetric eigenvector for a given eigenvalue: cross-product method.
// ---------------------------------------------------------------------------
__device__ inline void eigvec3(double M00, double M01, double M02,
                               double M11, double M12, double M22,
                               double lam, double v[3]) {
  double r0x = M00 - lam, r0y = M01,       r0z = M02;
  double r1x = M01,       r1y = M11 - lam, r1z = M12;
  double r2x = M02,       r2y = M12,       r2z = M22 - lam;
  double c0x = r0y*r1z - r0z*r1y, c0y = r0z*r1x - r0x*r1z, c0z = r0x*r1y - r0y*r1x;
  double c1x = r0y*r2z - r0z*r2y, c1y = r0z*r2x - r0x*r2z, c1z = r0x*r2y - r0y*r2x;
  double c2x = r1y*r2z - r1z*r2y, c2y = r1z*r2x - r1x*r2z, c2z = r1x*r2y - r1y*r2x;
  double n0 = c0x*c0x + c0y*c0y + c0z*c0z;
  double n1 = c1x*c1x + c1y*c1y + c1z*c1z;
  double n2 = c2x*c2x + c2y*c2y + c2z*c2z;
  double vx = c0x, vy = c0y, vz = c0z, n = n0;
  if (n1 > n) { vx = c1x; vy = c1y; vz = c1z; n = n1; }
  if (n2 > n) { vx = c2x; vy = c2y; vz = c2z; n = n2; }
  double inv = 1.0 / sqrt(n);
  v[0] = vx * inv; v[1] = vy * inv; v[2] = vz * inv;
}

// ---------------------------------------------------------------------------
// Kernel 1: per-batch center + WMMA covariance + closed-form eigh.
// One block (8 wave32) per batch.
//
// Phase 2 detail: C = Xc^T Xc via V_WMMA_F32_16X16X4_F32, 4 points/WMMA.
// A-matrix (16x4 f32) layout: lanes 0-15 hold K=0,1 in the two VGPRs,
// lanes 16-31 hold K=2,3; the 4x16 B layout coincides element-for-element,
// so one float2 per lane feeds both operands of the symmetric product.
// Only accumulator rows/cols 0-2 are ever read, so lanes with mrow>=3 may
// load garbage (any in-bounds address) -> branch-free unconditional
// ds_loads. 4 independent accumulators give ILP across the LDS->WMMA chain.
// ---------------------------------------------------------------------------
__global__ __launch_bounds__(256) void fa_stats_kernel(
    const float* __restrict__ X, const float* __restrict__ mask,
    float* __restrict__ out_F, float* __restrict__ out_center,
    float* __restrict__ ws) {
  __shared__ float sXc[NPTS * 3];  // 48 KB of the WGP's 320 KB LDS
  __shared__ float sred[4];
  __shared__ float sC[9];

  const int b   = blockIdx.x;
  const int tid = threadIdx.x;
  const float* __restrict__ Xb = X + (size_t)b * NPTS * 3;
  const float* __restrict__ Mb = mask + (size_t)b * NPTS;

  // ---- Phase 1: masked sums -> center --------------------------------------
  float sx = 0.f, sy = 0.f, sz = 0.f, sw = 0.f;
  for (int p = tid; p < NPTS; p += 256) {
    float m = Mb[p];
    sx += m * Xb[p * 3 + 0];
    sy += m * Xb[p * 3 + 1];
    sz += m * Xb[p * 3 + 2];
    sw += m;
  }
  if (tid < 4) sred[tid] = 0.f;
  if (tid < 9) sC[tid] = 0.f;
  __syncthreads();
  #pragma unroll
  for (int off = 16; off > 0; off >>= 1) {
    sx += __shfl_xor(sx, off);
    sy += __shfl_xor(sy, off);
    sz += __shfl_xor(sz, off);
    sw += __shfl_xor(sw, off);
  }
  if ((tid & 31) == 0) {
    atomicAdd(&sred[0], sx);
    atomicAdd(&sred[1], sy);
    atomicAdd(&sred[2], sz);
    atomicAdd(&sred[3], sw);
  }
  __syncthreads();
  const float invw = 1.0f / sred[3];
  const float cx = sred[0] * invw, cy = sred[1] * invw, cz = sred[2] * invw;

  // ---- Phase 1.5: Xc -> LDS (coalesced global reads) -----------------------
  for (int p = tid; p < NPTS; p += 256) {
    float m = Mb[p];
    sXc[p * 3 + 0] = Xb[p * 3 + 0] - cx * m;
    sXc[p * 3 + 1] = Xb[p * 3 + 1] - cy * m;
    sXc[p * 3 + 2] = Xb[p * 3 + 2] - cz * m;
  }
  __syncthreads();

  // ---- Phase 2: C = Xc^T Xc via WMMA, branch-free, 4-way unrolled ----------
  const int wave  = tid >> 5;   // 0..7
  const int lane  = tid & 31;
  const int mrow  = lane & 15;                 // M (coordinate) index
  const int khalf = lane >> 4;                 // 0: K=0,1   1: K=2,3
  const int mm    = (mrow < 3) ? mrow : 0;     // dead lanes: any in-bounds addr

  v8f acc0 = {}, acc1 = {}, acc2 = {}, acc3 = {};
  // groups of 4 points; wave handles g === wave (mod 8); 4 groups / iter
  for (int g = wave; g < NPTS / 4; g += 32) {
    const int q0 = (4 * g        + 2 * khalf) * 3 + mm;
    const int q1 = (4 * (g +  8) + 2 * khalf) * 3 + mm;
    const int q2 = (4 * (g + 16) + 2 * khalf) * 3 + mm;
    const int q3 = (4 * (g + 24) + 2 * khalf) * 3 + mm;
    v2f a0, a1, a2, a3;
    a0.x = sXc[q0]; a0.y = sXc[q0 + 3];
    a1.x = sXc[q1]; a1.y = sXc[q1 + 3];
    a2.x = sXc[q2]; a2.y = sXc[q2 + 3];
    a3.x = sXc[q3]; a3.y = sXc[q3 + 3];
    acc0 = __builtin_amdgcn_wmma_f32_16x16x4_f32(false, a0, false, a0,
                                                 (short)0, acc0, false, false);
    acc1 = __builtin_amdgcn_wmma_f32_16x16x4_f32(false, a1, false, a1,
                                                 (short)0, acc1, false, false);
    acc2 = __builtin_amdgcn_wmma_f32_16x16x4_f32(false, a2, false, a2,
                                                 (short)0, acc2, false, false);
    acc3 = __builtin_amdgcn_wmma_f32_16x16x4_f32(false, a3, false, a3,
                                                 (short)0, acc3, false, false);
  }
  // C/D layout: lane = column N (0-15), VGPR r = row M. Only 3x3 block read.
  if (lane < 3) {
    atomicAdd(&sC[0 * 3 + lane], acc0[0] + acc1[0] + acc2[0] + acc3[0]);
    atomicAdd(&sC[1 * 3 + lane], acc0[1] + acc1[1] + acc2[1] + acc3[1]);
    atomicAdd(&sC[2 * 3 + lane], acc0[2] + acc1[2] + acc2[2] + acc3[2]);
  }
  __syncthreads();

  // ---- Phase 3: eigh(C), write F_ops / center / workspace ------------------
  if (tid == 0) {
    double M00 = sC[0], M01 = sC[1], M02 = sC[2];
    double M11 = sC[4], M12 = sC[5], M22 = sC[8];

    double lam[3];
    double vec[3][3];  // vec[k][i] = component i of eigenvector k
    double q  = (M00 + M11 + M22) / 3.0;
    double p1 = M01 * M01 + M02 * M02 + M12 * M12;

    if (p1 <= 1e-14 * (q * q) + 1e-290) {
      double d[3] = {M00, M11, M22};
      int ord[3] = {0, 1, 2};
      if (d[ord[0]] > d[ord[1]]) { int t = ord[0]; ord[0] = ord[1]; ord[1] = t; }
      if (d[ord[1]] > d[ord[2]]) { int t = ord[1]; ord[1] = ord[2]; ord[2] = t; }
      if (d[ord[0]] > d[ord[1]]) { int t = ord[0]; ord[0] = ord[1]; ord[1] = t; }
      for (int k = 0; k < 3; ++k) {
        lam[k] = d[ord[k]];
        vec[k][0] = (ord[k] == 0) ? 1.0 : 0.0;
        vec[k][1] = (ord[k] == 1) ? 1.0 : 0.0;
        vec[k][2] = (ord[k] == 2) ? 1.0 : 0.0;
      }
    } else {
      double p2 = (M00 - q) * (M00 - q) + (M11 - q) * (M11 - q) +
                  (M22 - q) * (M22 - q) + 2.0 * p1;
      double p = sqrt(p2 / 6.0);
      double ip = 1.0 / p;
      double b00 = (M00 - q) * ip, b11 = (M11 - q) * ip, b22 = (M22 - q) * ip;
      double b01 = M01 * ip, b02 = M02 * ip, b12 = M12 * ip;
      double detB = b00 * (b11 * b22 - b12 * b12) -
                    b01 * (b01 * b22 - b12 * b02) +
                    b02 * (b01 * b12 - b11 * b02);
      double r = 0.5 * detB;
      r = fmin(1.0, fmax(-1.0, r));
      double phi = acos(r) / 3.0;
      double e2 = q + 2.0 * p * cos(phi);                       // largest
      double e0 = q + 2.0 * p * cos(phi + 2.0943951023931953);  // smallest
      double e1 = 3.0 * q - e0 - e2;
      lam[0] = e0; lam[1] = e1; lam[2] = e2;
      eigvec3(M00, M01, M02, M11, M12, M22, lam[0], vec[0]);
      eigvec3(M00, M01, M02, M11, M12, M22, lam[2], vec[2]);
      vec[1][0] = vec[2][1] * vec[0][2] - vec[2][2] * vec[0][1];
      vec[1][1] = vec[2][2] * vec[0][0] - vec[2][0] * vec[0][2];
      vec[1][2] = vec[2][0] * vec[0][1] - vec[2][1] * vec[0][0];
      double nn = vec[1][0] * vec[1][0] + vec[1][1] * vec[1][1] +
                  vec[1][2] * vec[1][2];
      double inv = 1.0 / sqrt(nn);
      vec[1][0] *= inv; vec[1][1] *= inv; vec[1][2] *= inv;
    }

    // deterministic sign: largest-|component| entry made positive
    for (int k = 0; k < 3; ++k) {
      double mbest = vec[k][0];
      if (fabs(vec[k][1]) > fabs(mbest)) mbest = vec[k][1];
      if (fabs(vec[k][2]) > fabs(mbest)) mbest = vec[k][2];
      if (mbest < 0.0) {
        vec[k][0] = -vec[k][0]; vec[k][1] = -vec[k][1]; vec[k][2] = -vec[k][2];
      }
    }

    // V[i][j] = component i of eigenvector j (eigh column convention)
    float Vf[3][3];
    for (int i = 0; i < 3; ++i)
      for (int j = 0; j < 3; ++j)
        Vf[i][j] = (float)vec[j][i];

    // F_ops[b,o,i,j] = OPS[o][j] * V[i][j]; sign bits: j=0->bit2, 1->bit1, 2->bit0
    float* Fb = out_F + (size_t)b * 72;
    for (int o = 0; o < 8; ++o) {
      float s[3];
      s[0] = (o & 4) ? 1.f : -1.f;
      s[1] = (o & 2) ? 1.f : -1.f;
      s[2] = (o & 1) ? 1.f : -1.f;
      for (int i = 0; i < 3; ++i)
        for (int j = 0; j < 3; ++j)
          Fb[o * 9 + i * 3 + j] = s[j] * Vf[i][j];
    }
    out_center[b * 3 + 0] = cx;
    out_center[b * 3 + 1] = cy;
    out_center[b * 3 + 2] = cz;

    float* wb = ws + (size_t)b * 12;
    for (int i = 0; i < 3; ++i)
      for (int j = 0; j < 3; ++j)
        wb[i * 3 + j] = Vf[i][j];
    wb[9] = cx; wb[10] = cy; wb[11] = cz;
  }
}

// ---------------------------------------------------------------------------
// Kernel 2: bandwidth-bound streaming kernel.
// h[b,o,p,i] = OPS[o][i] * (V^T Xc[p])_i : one eigenbasis projection per
// point, then 8 sign variants with coalesced 12B contiguous stores
// (~100 MB written + ~17 MB read -> ~5 us at 23.3 TB/s).
// ---------------------------------------------------------------------------
__global__ __launch_bounds__(256) void fa_apply_kernel(
    const float* __restrict__ X, const float* __restrict__ mask,
    const float* __restrict__ ws, float* __restrict__ out_h) {
  const int blocksPerBatch = NPTS / 256;  // 16
  const int b = blockIdx.x / blocksPerBatch;
  const int p = (blockIdx.x % blocksPerBatch) * 256 + threadIdx.x;

  // b is uniform per block -> scalar broadcasts of V and center
  const float* __restrict__ w = ws + (size_t)b * 12;
  const float V00 = w[0], V01 = w[1], V02 = w[2];
  const float V10 = w[3], V11 = w[4], V12 = w[5];
  const float V20 = w[6], V21 = w[7], V22 = w[8];
  const float cx = w[9], cy = w[10], cz = w[11];

  const size_t base = (size_t)b * NPTS + p;
  const float m = mask[base];
  const float x = X[base * 3 + 0] - cx * m;
  const float y = X[base * 3 + 1] - cy * m;
  const float z = X[base * 3 + 2] - cz * m;

  const float gx = V00 * x + V10 * y + V20 * z;
  const float gy = V01 * x + V11 * y + V21 * z;
  const float gz = V02 * x + V12 * y + V22 * z;

  float* __restrict__ hb = out_h + (size_t)b * 8 * NPTS * 3;
  #pragma unroll
  for (int o = 0; o < 8; ++o) {
    const float s0 = (o & 4) ? 1.f : -1.f;
    const float s1 = (o & 2) ? 1.f : -1.f;
    const float s2 = (o & 1) ? 1.f : -1.f;
    const size_t off = ((size_t)o * NPTS + p) * 3;
    hb[off + 0] = s0 * gx;
    hb[off + 1] = s1 * gy;
    hb[off + 2] = s2 * gz;
  }
}

// ---------------------------------------------------------------------------
// Launch: outputs concatenated as (h, F_ops, center).
// ---------------------------------------------------------------------------
extern "C" void kernel_launch(void* const* d_in, const int* in_sizes, int n_in,
                              void* d_out, int out_size, void* d_ws, size_t ws_size,
                              hipStream_t stream) {
  const float* X    = (const float*)d_in[0];
  const float* mask = (const float*)d_in[1];
  float* out = (float*)d_out;

  const size_t H_ELEMS = (size_t)NB * 8 * NPTS * 3;  // 25,165,824
  const size_t F_ELEMS = (size_t)NB * 72;            // 18,432
  float* out_h = out;
  float* out_F = out + H_ELEMS;
  float* out_c = out + H_ELEMS + F_ELEMS;
  float* wsf   = (float*)d_ws;  // NB*12 floats

  fa_stats_kernel<<<NB, 256, 0, stream>>>(X, mask, out_F, out_c, wsf);
  fa_apply_kernel<<<NB * (NPTS / 256), 256, 0, stream>>>(X, mask, wsf, out_h);
}